// loss_SAW_2_83863531421940
// MI455X (gfx1250) — compile-verified
//
#include <hip/hip_runtime.h>

#define EPSF      1e-6f
#define MAXB      65          // numMask(64) + background
#define HIST_PAD  72          // pad bins to spread LDS banks
#define NWAVES    8           // 256 threads / wave32
#define TPB       256
#define VEC       4           // floats per thread per stage (16B -> b128)
#define STAGE_ELEMS (TPB * VEC)   // 1024 elements per stage per workgroup

// ---- probe for gfx1250 async global->LDS DMA builtins -----------------------
#if defined(__has_builtin)
#if __has_builtin(__builtin_amdgcn_global_load_async_to_lds_b128) && \
    __has_builtin(__builtin_amdgcn_s_wait_asynccnt)
#define USE_ASYNC_LDS 1
#endif
#endif
#ifndef USE_ASYNC_LDS
#define USE_ASYNC_LDS 0
#endif

typedef int v4i __attribute__((vector_size(16)));

#if USE_ASYNC_LDS
__device__ __forceinline__ void async_copy16(const void* gsrc, void* ldst) {
  // 16B per lane: copies 16 bytes from the global address gsrc to the LDS address ldst   (tracked by ASYNCcnt)
  __builtin_amdgcn_global_load_async_to_lds_b128(
      (__attribute__((address_space(1))) v4i*)gsrc,
      (__attribute__((address_space(3))) v4i*)ldst,
      /*offset=*/0, /*cpol=*/0);
}
#endif

// ---- stage 1: per-workgroup partial (count,sum,sumsq) histograms ------------
__global__ void __launch_bounds__(TPB)
saw_partial_kernel(const float* __restrict__ shading,
                   const int*   __restrict__ mask,
                   int n,
                   float* __restrict__ ws) {
  __shared__ __align__(16) float s_sh[2][STAGE_ELEMS];
  __shared__ __align__(16) int   s_mk[2][STAGE_ELEMS];
  __shared__ float s_hist[NWAVES][3][HIST_PAD];   // [wave][sum|sumsq|cnt][bin]

  const int tid = threadIdx.x;
  const int wv  = tid >> 5;

  for (int i = tid; i < NWAVES * 3 * HIST_PAD; i += TPB)
    ((float*)s_hist)[i] = 0.0f;
  __syncthreads();

  float* hsum = s_hist[wv][0];
  float* hsq  = s_hist[wv][1];
  float* hcnt = s_hist[wv][2];

  auto accum = [&](float v, int mi) {
    unsigned b = (unsigned)mi;
    if (b < (unsigned)MAXB) {
      atomicAdd(&hsum[b], v);        // ds_add_f32 (no-rtn)
      atomicAdd(&hsq[b],  v * v);
      atomicAdd(&hcnt[b], 1.0f);
    }
  };

  const int n_stages = n / STAGE_ELEMS;
  const int n_full   = n_stages * STAGE_ELEMS;

#if USE_ASYNC_LDS
  // double-buffered async pipeline: copy tile k+1 while histogramming tile k
  long long t = blockIdx.x;
  int buf = 0;
  if (t < (long long)n_stages) {
    size_t base = (size_t)t * STAGE_ELEMS + (size_t)tid * VEC;
    async_copy16(shading + base, &s_sh[0][tid * VEC]);
    async_copy16(mask    + base, &s_mk[0][tid * VEC]);
  }
  for (; t < (long long)n_stages; t += gridDim.x) {
    long long tn = t + gridDim.x;
    if (tn < (long long)n_stages) {
      size_t base = (size_t)tn * STAGE_ELEMS + (size_t)tid * VEC;
      async_copy16(shading + base, &s_sh[buf ^ 1][tid * VEC]);
      async_copy16(mask    + base, &s_mk[buf ^ 1][tid * VEC]);
      __builtin_amdgcn_s_wait_asynccnt(2);   // current tile's 2 copies done
    } else {
      __builtin_amdgcn_s_wait_asynccnt(0);
    }
    __asm__ volatile("" ::: "memory");
    // each lane consumes exactly the 16B it copied -> no barrier needed
    float4 v = *reinterpret_cast<const float4*>(&s_sh[buf][tid * VEC]);
    int4   m = *reinterpret_cast<const int4*>(&s_mk[buf][tid * VEC]);
    accum(v.x, m.x); accum(v.y, m.y); accum(v.z, m.z); accum(v.w, m.w);
    buf ^= 1;
  }
#else
  for (long long t = blockIdx.x; t < (long long)n_stages; t += gridDim.x) {
    size_t base = (size_t)t * STAGE_ELEMS + (size_t)tid * VEC;
    __builtin_prefetch(shading + base + (size_t)gridDim.x * STAGE_ELEMS, 0, 1);
    float4 v = *reinterpret_cast<const float4*>(shading + base);
    int4   m = *reinterpret_cast<const int4*>(mask + base);
    accum(v.x, m.x); accum(v.y, m.y); accum(v.z, m.z); accum(v.w, m.w);
  }
#endif

  // tail (n not multiple of STAGE_ELEMS) handled by block 0
  if (blockIdx.x == 0) {
    for (long long i = (long long)n_full + tid; i < (long long)n; i += TPB)
      accum(shading[i], mask[i]);
  }

  __syncthreads();
  // fold the 8 per-wave histograms and store this workgroup's partials
  for (int idx = tid; idx < 3 * MAXB; idx += TPB) {
    int comp = idx / MAXB;
    int b    = idx % MAXB;
    float acc = 0.0f;
    for (int w = 0; w < NWAVES; ++w) acc += s_hist[w][comp][b];
    ws[(size_t)blockIdx.x * (3 * MAXB) + idx] = acc;
  }
}

// ---- stage 2: deterministic reduction of partials + f32 epilogue ------------
__global__ void __launch_bounds__(TPB)
saw_final_kernel(const float* __restrict__ ws, int G,
                 const int* __restrict__ nm_ptr,
                 float* __restrict__ out, int n) {
  __shared__ float fin[3 * MAXB];
  const int tid = threadIdx.x;

  for (int idx = tid; idx < 3 * MAXB; idx += TPB) {
    float acc = 0.0f;
    for (int g = 0; g < G; ++g)           // fixed order -> deterministic
      acc += ws[(size_t)g * (3 * MAXB) + idx];
    fin[idx] = acc;
  }
  __syncthreads();

  if (tid == 0) {
    int numMask = *nm_ptr;
    int nseg = numMask + 1;
    if (nseg > MAXB) nseg = MAXB;
    const float num = (float)n;           // H*W, matches reference f32 math
    float total = 0.0f;
    for (int b = 1; b < nseg; ++b) {
      float sum = fin[0 * MAXB + b];
      float sq  = fin[1 * MAXB + b];
      float cnt = fin[2 * MAXB + b];
      float N   = cnt * num;
      float s1  = sq / (N + EPSF);
      float s2  = (sum * sum) / (N * N + EPSF);
      total += s1 - s2;
    }
    out[0] = total / ((float)numMask + EPSF);
  }
}

extern "C" void kernel_launch(void* const* d_in, const int* in_sizes, int n_in,
                              void* d_out, int out_size, void* d_ws, size_t ws_size,
                              hipStream_t stream) {
  const float* shading = (const float*)d_in[0];
  const int*   mask    = (const int*)d_in[1];
  const int*   nm_ptr  = (const int*)d_in[2];
  const int n = in_sizes[0];

  float* ws = (float*)d_ws;
  int G = 1024;
  size_t need = (size_t)G * (3 * MAXB) * sizeof(float);
  if (need > ws_size) {
    G = (int)(ws_size / ((3 * MAXB) * sizeof(float)));
    if (G < 1) G = 1;
  }

  saw_partial_kernel<<<G, TPB, 0, stream>>>(shading, mask, n, ws);
  saw_final_kernel<<<1, TPB, 0, stream>>>(ws, G, nm_ptr, (float*)d_out, n);
}